// SpatialTransformer_71201967833644
// MI455X (gfx1250) — compile-verified
//
#include <hip/hip_runtime.h>
#include <hip/hip_bf16.h>
#include <math.h>

typedef __attribute__((ext_vector_type(2))) float v2f;
typedef __attribute__((ext_vector_type(8))) float v8f;

#define WAVES_PER_BLOCK 8

static __device__ __forceinline__ v8f wmma_f32_16x16x4(v2f a, v2f b, v8f c) {
  // 8 args: (neg_a, A, neg_b, B, c_mod, C, reuse_a, reuse_b)
  return __builtin_amdgcn_wmma_f32_16x16x4_f32(false, a, false, b, (short)0, c,
                                               false, false);
}

// ---------------------------------------------------------------------------
// Kernel 1: QKV projection.  x[32768,64] @ w_qkv[64,48] -> q/k/v in [b,h,n,4]
// One wave per 16-row tile; 3 N-tiles of 16; K loop = 16 chunks of 4.
// ---------------------------------------------------------------------------
__global__ __launch_bounds__(256) void qkv_proj_16x48(
    const float* __restrict__ x, const float* __restrict__ wqkv,
    float* __restrict__ q, float* __restrict__ k, float* __restrict__ v) {
  const int lane = threadIdx.x & 31;
  const int wave = threadIdx.x >> 5;
  const int half = lane >> 4;   // upper half-wave
  const int l16  = lane & 15;
  const int mtile = blockIdx.x * WAVES_PER_BLOCK + wave;  // 0..2047

  const v8f vzero = {0.f, 0.f, 0.f, 0.f, 0.f, 0.f, 0.f, 0.f};
  v8f acc[3];
#pragma unroll
  for (int i = 0; i < 3; ++i) acc[i] = vzero;

  const float* xrow = x + (mtile * 16 + l16) * 64;  // A: row = l16
#pragma unroll 4
  for (int kc = 0; kc < 16; ++kc) {
    const int kb = kc * 4 + 2 * half;  // A/B K index base: K = kb + j
    v2f a;
    a.x = xrow[kb + 0];
    a.y = xrow[kb + 1];
#pragma unroll
    for (int nt = 0; nt < 3; ++nt) {
      const int nn = nt * 16 + l16;  // B: col = l16
      v2f b;
      b.x = wqkv[(kb + 0) * 48 + nn];
      b.y = wqkv[(kb + 1) * 48 + nn];
      acc[nt] = wmma_f32_16x16x4(a, b, acc[nt]);
    }
  }

  // D layout: reg r, lanes 0-15 -> row r, lanes 16-31 -> row r+8; col = l16.
  // N-tile 0/1/2 == q/k/v exactly (inner == 16). col -> (h, d).
  const int h = l16 >> 2, d = l16 & 3;
#pragma unroll
  for (int nt = 0; nt < 3; ++nt) {
    float* dst = (nt == 0) ? q : ((nt == 1) ? k : v);
#pragma unroll
    for (int r = 0; r < 8; ++r) {
      const int row = mtile * 16 + r + 8 * half;  // global b*n row
      const int bb = row >> 11, nn = row & 2047;
      dst[(((bb << 2) + h) * 2048 + nn) * 4 + d] = acc[nt][r];
    }
  }
}

// ---------------------------------------------------------------------------
// Kernel 2: streaming (flash) attention, one wave per (b, h, 16-query tile).
// S tile via WMMA f32 16x16x4 (K = DIM_HEAD = 4); online max via half-wave
// shuffle reductions; the softmax denominator is accumulated BY the P@V WMMA
// through a ones-column (B column 4), so no row-sum shuffles are needed.
// P is routed through a per-wave LDS slab to build A-fragments for P@V.
// ---------------------------------------------------------------------------
__global__ __launch_bounds__(256) void attn_fwd(
    const float* __restrict__ q, const float* __restrict__ k,
    const float* __restrict__ v, float* __restrict__ o /* [b,n,16] merged */) {
  __shared__ float pbuf[WAVES_PER_BLOCK][16][18];  // stride 18: b64-aligned reads

  const int lane = threadIdx.x & 31;
  const int wave = threadIdx.x >> 5;
  const int half = lane >> 4;
  const int l16  = lane & 15;

  const int wid = blockIdx.x * WAVES_PER_BLOCK + wave;  // 0..8191
  const int qt = wid & 127;   // query tile
  const int bh = wid >> 7;    // fused (b,h)
  const int h  = bh & 3;
  const int bb = bh >> 2;

  const float* qb = q + bh * 2048 * 4;
  const float* kb = k + bh * 2048 * 4;
  const float* vb = v + bh * 2048 * 4;

  // Q A-fragment (constant across key loop), SCALE = 1/sqrt(4) folded in.
  v2f aq;
  {
    const int qrow = qt * 16 + l16;
    aq.x = qb[qrow * 4 + 2 * half + 0] * 0.5f;
    aq.y = qb[qrow * 4 + 2 * half + 1] * 0.5f;
  }

  const v8f vzero = {0.f, 0.f, 0.f, 0.f, 0.f, 0.f, 0.f, 0.f};
  v8f acc = vzero;  // D layout: cols 0-3 = head dims, col 4 = softmax denom l
  float m[8];       // per-row running max (valid per half-wave)
#pragma unroll
  for (int r = 0; r < 8; ++r) m[r] = -3.0e38f;

  const float* prow = &pbuf[wave][l16][0];  // A-fragment read base (row = l16)

  for (int kt = 0; kt < 128; ++kt) {
    // K^T B-fragment: B[d][key], d = j + 2*half, key(col) = kt*16 + l16
    v2f bk;
    const int krow = kt * 16 + l16;
    bk.x = kb[krow * 4 + 2 * half + 0];
    bk.y = kb[krow * 4 + 2 * half + 1];
    v8f s = wmma_f32_16x16x4(aq, bk, vzero);  // S[q][key], pre-scaled

    // Online max + exp per row (rows striped: reg r -> row r + 8*half).
#pragma unroll
    for (int r = 0; r < 8; ++r) {
      const float sv = s[r];
      float rm = sv;
      rm = fmaxf(rm, __shfl_xor(rm, 1, 32));
      rm = fmaxf(rm, __shfl_xor(rm, 2, 32));
      rm = fmaxf(rm, __shfl_xor(rm, 4, 32));
      rm = fmaxf(rm, __shfl_xor(rm, 8, 32));
      const float mnew  = fmaxf(m[r], rm);
      const float alpha = __expf(m[r] - mnew);
      m[r] = mnew;
      acc[r] = acc[r] * alpha;  // rescales dims AND the col-4 denominator
      pbuf[wave][r + 8 * half][l16] = __expf(sv - mnew);
    }

    // P @ [V | 1] : 4 WMMAs, K-chunks of 4 keys. B col 4 = 1.0 accumulates
    // the row-sum of P (softmax denominator) directly into acc col 4.
#pragma unroll
    for (int c = 0; c < 4; ++c) {
      v2f ap;  // A: row = l16 (query), K = c*4 + 2*half + j
      ap.x = prow[c * 4 + 2 * half + 0];
      ap.y = prow[c * 4 + 2 * half + 1];
      v2f bv;  // B: row = key in chunk, col = head dim / ones-column / zero
      const int key0 = kt * 16 + c * 4 + 2 * half;
      bv.x = (l16 < 4) ? vb[(key0 + 0) * 4 + l16] : ((l16 == 4) ? 1.0f : 0.0f);
      bv.y = (l16 < 4) ? vb[(key0 + 1) * 4 + l16] : ((l16 == 4) ? 1.0f : 0.0f);
      acc = wmma_f32_16x16x4(ap, bv, acc);
    }
  }

  // Normalize by the WMMA-accumulated denominator (col 4 of each row) and
  // store merged 'b n (h d)' layout. Only cols 0-3 carry head dims.
#pragma unroll
  for (int r = 0; r < 8; ++r) {
    const float lr = __shfl(acc[r], (half << 4) + 4, 32);  // col-4 of this half
    if (l16 < 4) {
      const int row = qt * 16 + r + 8 * half;
      o[(bb * 2048 + row) * 16 + h * 4 + l16] = acc[r] / lr;
    }
  }
}

// ---------------------------------------------------------------------------
// Kernel 3: output projection. a[32768,16] @ w_out[16,64] + b_out
// ---------------------------------------------------------------------------
__global__ __launch_bounds__(256) void out_proj_16x64(
    const float* __restrict__ a, const float* __restrict__ w,
    const float* __restrict__ bias, float* __restrict__ out) {
  const int lane = threadIdx.x & 31;
  const int wave = threadIdx.x >> 5;
  const int half = lane >> 4;
  const int l16  = lane & 15;
  const int mtile = blockIdx.x * WAVES_PER_BLOCK + wave;  // 0..2047

  const v8f vzero = {0.f, 0.f, 0.f, 0.f, 0.f, 0.f, 0.f, 0.f};
  v8f acc[4];
#pragma unroll
  for (int i = 0; i < 4; ++i) acc[i] = vzero;

  const float* arow = a + (mtile * 16 + l16) * 16;
#pragma unroll
  for (int kc = 0; kc < 4; ++kc) {
    const int kb = kc * 4 + 2 * half;
    v2f af;
    af.x = arow[kb + 0];
    af.y = arow[kb + 1];
#pragma unroll
    for (int nt = 0; nt < 4; ++nt) {
      const int nn = nt * 16 + l16;
      v2f bf;
      bf.x = w[(kb + 0) * 64 + nn];
      bf.y = w[(kb + 1) * 64 + nn];
      acc[nt] = wmma_f32_16x16x4(af, bf, acc[nt]);
    }
  }

#pragma unroll
  for (int nt = 0; nt < 4; ++nt) {
    const int nn = nt * 16 + l16;
    const float bo = bias[nn];
#pragma unroll
    for (int r = 0; r < 8; ++r) {
      const int row = mtile * 16 + r + 8 * half;
      out[row * 64 + nn] = acc[nt][r] + bo;
    }
  }
}

// ---------------------------------------------------------------------------
extern "C" void kernel_launch(void* const* d_in, const int* in_sizes, int n_in,
                              void* d_out, int out_size, void* d_ws,
                              size_t ws_size, hipStream_t stream) {
  (void)in_sizes; (void)n_in; (void)out_size; (void)ws_size;

  const float* x    = (const float*)d_in[0];  // [16,2048,64]
  const float* wqkv = (const float*)d_in[1];  // [64,48]
  const float* wout = (const float*)d_in[2];  // [16,64]
  const float* bout = (const float*)d_in[3];  // [64]
  float* out = (float*)d_out;                 // [16,2048,64]

  const size_t kQkvElems = (size_t)16 * 4 * 2048 * 4;  // 524288 floats (2 MB)
  float* qw = (float*)d_ws;
  float* kw = qw + kQkvElems;
  float* vw = kw + kQkvElems;
  float* aw = vw + kQkvElems;  // merged attention out [16,2048,16]

  dim3 blk(256);
  qkv_proj_16x48<<<2048 / WAVES_PER_BLOCK, blk, 0, stream>>>(x, wqkv, qw, kw, vw);
  attn_fwd<<<8192 / WAVES_PER_BLOCK, blk, 0, stream>>>(qw, kw, vw, aw);
  out_proj_16x64<<<2048 / WAVES_PER_BLOCK, blk, 0, stream>>>(aw, wout, bout, out);
}